// CompGCNConv_83640193122546
// MI455X (gfx1250) — compile-verified
//
#include <hip/hip_runtime.h>
#include <math.h>

typedef float v2f __attribute__((ext_vector_type(2)));
typedef float v8f __attribute__((ext_vector_type(8)));

#define DIM 128
#define LDS_STRIDE 132   // 128 + 4 pad: keeps float4 alignment, spreads LDS banks

// Build packed-rFFT matrix F [128 time x 128 packed] and packed-irFFT matrix
// G [128 packed x 128 time]. Packing: slot k (0..64) = Re[k], slot 64+k
// (k=1..63) = Im[k]. Exact angle reduction via (j*k) mod 128.
__global__ void build_dft(float* __restrict__ F, float* __restrict__ G) {
    const int j = blockIdx.x;   // 0..127
    const int c = threadIdx.x;  // 0..127
    const float w = 6.28318530717958647692f / 128.0f;

    float fv;
    if (c <= 64) {
        int m = (j * c) & 127;
        fv = cosf(w * (float)m);
    } else {
        int k = c - 64;
        int m = (j * k) & 127;
        fv = -sinf(w * (float)m);
    }
    F[j * DIM + c] = fv;

    const int p = j, t = c;
    float gv;
    if (p == 0) {
        gv = 1.0f / 128.0f;
    } else if (p == 64) {
        gv = ((t & 1) ? -1.0f : 1.0f) / 128.0f;
    } else if (p < 64) {
        int m = (p * t) & 127;
        gv = 2.0f * cosf(w * (float)m) / 128.0f;
    } else {
        int k = p - 64;
        int m = (k * t) & 127;
        gv = -2.0f * sinf(w * (float)m) / 128.0f;
    }
    G[p * DIM + t] = gv;
}

// Pack B [128x128] into fragment-interleaved Bp: Bp[(p*128 + col)*2 + j] = B[2p+j][col].
// A wave's B fragment for k-step ks is then ONE contiguous float2 at (2*ks+half)*128+col.
__global__ void pack_b_pairs(const float* __restrict__ B, float* __restrict__ Bp) {
    const int col = threadIdx.x;  // 0..127
    const int p   = blockIdx.x;   // 0..63
    Bp[(p * DIM + col) * 2 + 0] = B[(2 * p + 0) * DIM + col];
    Bp[(p * DIM + col) * 2 + 1] = B[(2 * p + 1) * DIM + col];
}

// C[M x 128] = A[M x 128] @ B[128 x 128] (+ bias), fp32 via V_WMMA_F32_16X16X4_F32.
// B supplied in packed-pair form (pack_b_pairs). A tile staged through LDS with a
// single hoisted bounds check (rows >= a_valid read as zero). Safe for C==A.
__global__ void gemm128_wmma(const float* __restrict__ A,
                             const float* __restrict__ Bp,
                             float* __restrict__ C,
                             int M, int a_valid,
                             const float* __restrict__ bias) {
    __shared__ float As[16 * LDS_STRIDE];
    const int tid = threadIdx.x;
    const int row_base = blockIdx.x * 16;

    // Cooperative stage: 256 threads x 8 floats = 16x128 tile, b128 global loads.
    {
        const int r    = tid >> 4;          // 0..15
        const int cc   = (tid & 15) * 8;    // 0..120
        const int grow = row_base + r;
        float4 v0 = make_float4(0.f, 0.f, 0.f, 0.f);
        float4 v1 = v0;
        if (grow < a_valid) {
            const float4* Ap4 = (const float4*)(A + (size_t)grow * DIM + cc);
            v0 = Ap4[0];
            v1 = Ap4[1];
        }
        float4* s4 = (float4*)(As + r * LDS_STRIDE + cc);
        s4[0] = v0;
        s4[1] = v1;
    }
    __syncthreads();

    const int lane = tid & 31;
    const int wave = tid >> 5;               // column tile 0..7
    const int half = lane >> 4;              // 0: K pair {0,1}, 1: {2,3}
    const int col  = wave * 16 + (lane & 15);
    const float2* As2 = (const float2*)As;
    const float2* Bp2 = (const float2*)Bp;
    const int arow2 = (lane & 15) * (LDS_STRIDE / 2);

    v8f acc = {0.f, 0.f, 0.f, 0.f, 0.f, 0.f, 0.f, 0.f};

#pragma unroll
    for (int ks = 0; ks < 32; ++ks) {
        // A 16x4 fragment (ISA 7.12.2): lane<16 -> K={4ks,4ks+1}, lane>=16 -> K={4ks+2,4ks+3}
        const float2 af = As2[arow2 + ks * 2 + half];          // ds_load_b64
        const float2 bf = Bp2[(size_t)(2 * ks + half) * DIM + col]; // global_load_b64
        v2f a; a.x = af.x; a.y = af.y;
        v2f b; b.x = bf.x; b.y = bf.y;
        acc = __builtin_amdgcn_wmma_f32_16x16x4_f32(
            /*neg_a=*/false, a, /*neg_b=*/false, b,
            /*c_mod=*/(short)0, acc, /*reuse_a=*/false, /*reuse_b=*/false);
    }

    const float bv = bias ? bias[col] : 0.0f;
    if (row_base + 16 <= M) {
        // Fast path: full tile (always taken for the 50000-row GEMMs).
#pragma unroll
        for (int i = 0; i < 8; ++i) {
            C[(size_t)(row_base + i + 8 * half) * DIM + col] = acc[i] + bv;
        }
    } else {
#pragma unroll
        for (int i = 0; i < 8; ++i) {
            const int row = row_base + i + 8 * half;
            if (row < M) C[(size_t)row * DIM + col] = acc[i] + bv;
        }
    }
}

__global__ void zero_f32(float* __restrict__ p, long long n) {
    long long i = (long long)blockIdx.x * blockDim.x + threadIdx.x;
    if (i < n) p[i] = 0.0f;
}

// One edge per wave: packed spectrum product conj(xf[src]) * relf[et],
// atomic scatter-add into acc[dst]. Lane l handles slots {2l,2l+1,64+2l,64+2l+1}.
__global__ void edge_scatter(const int* __restrict__ ei, const int* __restrict__ et,
                             const float* __restrict__ xf, const float* __restrict__ relf,
                             float* __restrict__ acc, int n_edges) {
    const int lane = threadIdx.x & 31;
    const int e = blockIdx.x * (blockDim.x >> 5) + (threadIdx.x >> 5);
    if (e >= n_edges) return;

    const int src = ei[e];
    const int dst = ei[n_edges + e];
    const int t   = et[e];

    const float2* xa = (const float2*)(xf + (size_t)src * DIM);
    const float2* xb = (const float2*)(relf + (size_t)t * DIM);
    const float2 alo = xa[lane];        // Re bins 2l, 2l+1
    const float2 ahi = xa[32 + lane];   // Im bins 2l, 2l+1 (lane0: slot64 = Re[64])
    const float2 blo = xb[lane];
    const float2 bhi = xb[32 + lane];

    // conj(a)*b : re = ar*br + ai*bi ; im = ar*bi - ai*br
    float olox = alo.x * blo.x + ahi.x * bhi.x;
    float oloy = alo.y * blo.y + ahi.y * bhi.y;
    float ohix = alo.x * bhi.x - ahi.x * blo.x;
    float ohiy = alo.y * bhi.y - ahi.y * blo.y;
    if (lane == 0) {                    // purely-real bins k=0 and k=64
        olox = alo.x * blo.x;           // bin 0
        ohix = ahi.x * bhi.x;           // bin 64 (stored at slot 64)
    }

    float* o = acc + (size_t)dst * DIM;
    const int c = 2 * lane;
    unsafeAtomicAdd(o + c,          olox);
    unsafeAtomicAdd(o + c + 1,      oloy);
    unsafeAtomicAdd(o + 64 + c,     ohix);
    unsafeAtomicAdd(o + 64 + c + 1, ohiy);
}

extern "C" void kernel_launch(void* const* d_in, const int* in_sizes, int n_in,
                              void* d_out, int out_size, void* d_ws, size_t ws_size,
                              hipStream_t stream) {
    const float* x          = (const float*)d_in[0];
    const int*   ei         = (const int*)  d_in[1];
    const int*   et         = (const int*)  d_in[2];
    const float* rel        = (const float*)d_in[3];
    const float* weight     = (const float*)d_in[4];
    const float* rel_weight = (const float*)d_in[5];
    const float* bias       = (const float*)d_in[6];

    const int n_nodes      = in_sizes[0] / DIM;   // 50000
    const int n_edges      = in_sizes[2];         // 600000
    const int rel_rows     = in_sizes[3] / DIM;   // 400
    const int rel_out_rows = rel_rows + 1;        // 401 (appended zero row)

    // Workspace layout (floats)
    float* F    = (float*)d_ws;                       // 128x128
    float* G    = F   + DIM * DIM;                    // 128x128
    float* Mm   = G   + DIM * DIM;                    // 128x128  (= G @ weight)
    float* Fp   = Mm  + DIM * DIM;                    // packed F
    float* Wp   = Fp  + DIM * DIM;                    // packed weight
    float* RWp  = Wp  + DIM * DIM;                    // packed rel_weight
    float* Mmp  = RWp + DIM * DIM;                    // packed Mm
    float* relf = Mmp + DIM * DIM;                    // 401x128
    float* xf   = relf + (size_t)rel_out_rows * DIM;  // 50000x128

    float* out     = (float*)d_out;                   // acc (freq) then final out, in place
    float* rel_out = out + (size_t)n_nodes * DIM;

    // 1) DFT/IDFT matrices + packed B operands
    build_dft<<<DIM, DIM, 0, stream>>>(F, G);
    pack_b_pairs<<<DIM / 2, DIM, 0, stream>>>(F, Fp);
    pack_b_pairs<<<DIM / 2, DIM, 0, stream>>>(weight, Wp);
    pack_b_pairs<<<DIM / 2, DIM, 0, stream>>>(rel_weight, RWp);

    // 2) Spectra: xf = x @ F ; relf = [rel;0] @ F
    gemm128_wmma<<<(n_nodes + 15) / 16, 256, 0, stream>>>(x, Fp, xf, n_nodes, n_nodes, nullptr);
    gemm128_wmma<<<(rel_out_rows + 15) / 16, 256, 0, stream>>>(rel, Fp, relf, rel_out_rows, rel_rows, nullptr);

    // 3) rel_out = [rel;0] @ rel_weight
    gemm128_wmma<<<(rel_out_rows + 15) / 16, 256, 0, stream>>>(rel, RWp, rel_out, rel_out_rows, rel_rows, nullptr);

    // 4) Fused inverse-FFT + projection matrix: Mm = G @ weight, then pack it
    gemm128_wmma<<<(DIM + 15) / 16, 256, 0, stream>>>(G, Wp, Mm, DIM, DIM, nullptr);
    pack_b_pairs<<<DIM / 2, DIM, 0, stream>>>(Mm, Mmp);

    // 5) acc = 0 ; frequency-domain scatter (self-loops have zero rel row -> skipped)
    const long long accN = (long long)n_nodes * DIM;
    zero_f32<<<(int)((accN + 255) / 256), 256, 0, stream>>>(out, accN);
    edge_scatter<<<(n_edges + 7) / 8, 256, 0, stream>>>(ei, et, xf, relf, out, n_edges);

    // 6) out = acc @ Mm + bias   (in-place per 16-row tile)
    gemm128_wmma<<<(n_nodes + 15) / 16, 256, 0, stream>>>(out, Mmp, out, n_nodes, n_nodes, bias);
}